// BBBP_GCN_attr_60404420051514
// MI455X (gfx1250) — compile-verified
//
#include <hip/hip_runtime.h>

typedef __attribute__((ext_vector_type(2))) float v2f;
typedef __attribute__((ext_vector_type(8))) float v8f;

#define H 128
#define TILE_M 64

// ---------------------------------------------------------------------------
// Dense GEMM  Y[nrows,128] = X[nrows,128] @ W[128,128]  using fp32 WMMA.
// Block = 256 threads (8 waves). Block computes a 64x128 slab.
// Wave w owns column tile n0 = 16*w. 4 row tiles of 16 per wave.
// ---------------------------------------------------------------------------
__global__ __launch_bounds__(256) void k_gemm128_wmma(
    const float* __restrict__ X, const float* __restrict__ W,
    float* __restrict__ Y, int nrows)
{
    __shared__ float As[TILE_M][130];   // +2 pad: kills 16-way bank conflicts

    const int tid  = threadIdx.x;
    const int wave = tid >> 5;
    const int lane = tid & 31;
    const int half = lane >> 4;         // 0: lanes 0-15, 1: lanes 16-31
    const int lm   = lane & 15;
    const long row0 = (long)blockIdx.x * TILE_M;

    // Stage A tile (64 x 128), zero-padding rows past nrows.
    for (int i = tid; i < TILE_M * 32; i += 256) {
        int r  = i >> 5;          // tile row
        int c4 = i & 31;          // float4 column chunk
        long gr = row0 + r;
        float4 v = make_float4(0.f, 0.f, 0.f, 0.f);
        if (gr < nrows) v = *(const float4*)(X + gr * H + c4 * 4);
        float* p = &As[r][c4 * 4];
        p[0] = v.x; p[1] = v.y; p[2] = v.z; p[3] = v.w;
    }
    __syncthreads();

    const int n0 = wave * 16;

    // Preload B fragments for all 32 K-steps: v_r = W[4t + 2*half + r][n0+lm]
    float2 bf[32];
#pragma unroll
    for (int t = 0; t < 32; ++t) {
        int k = 4 * t + 2 * half;
        bf[t].x = W[(long)k * H + n0 + lm];
        bf[t].y = W[(long)(k + 1) * H + n0 + lm];
    }

    v8f acc[4] = {};

#pragma unroll
    for (int t = 0; t < 32; ++t) {
        int k = 4 * t + 2 * half;
        v2f b; b.x = bf[t].x; b.y = bf[t].y;
#pragma unroll
        for (int mt = 0; mt < 4; ++mt) {
            // A fragment: lane holds A[m][2*half], A[m][2*half+1] (8B aligned)
            const float2 av = *(const float2*)&As[mt * 16 + lm][k];
            v2f a; a.x = av.x; a.y = av.y;
            acc[mt] = __builtin_amdgcn_wmma_f32_16x16x4_f32(
                false, a, false, b, (short)0, acc[mt], false, false);
        }
    }

    // C/D layout: VGPR r -> M = r + 8*half, N = lm
#pragma unroll
    for (int mt = 0; mt < 4; ++mt) {
#pragma unroll
        for (int r = 0; r < 8; ++r) {
            long gr = row0 + mt * 16 + half * 8 + r;
            if (gr < nrows) Y[gr * H + n0 + lm] = acc[mt][r];
        }
    }
}

// ---------------------------------------------------------------------------
// gcn_norm helpers
// ---------------------------------------------------------------------------
__global__ void k_fill1(float* __restrict__ p, int n) {
    int i = blockIdx.x * blockDim.x + threadIdx.x;
    if (i < n) p[i] = 1.0f;                     // self-loop weight
}

__global__ void k_deg_accum(const int* __restrict__ dst,
                            const float* __restrict__ w,
                            float* __restrict__ deg, int E) {
    int e = blockIdx.x * blockDim.x + threadIdx.x;
    if (e < E) atomicAdd(&deg[dst[e]], w[e]);
}

__global__ void k_dinv(const float* __restrict__ deg,
                       float* __restrict__ dinv, int n) {
    int i = blockIdx.x * blockDim.x + threadIdx.x;
    if (i < n) { float d = deg[i]; dinv[i] = (d > 0.f) ? rsqrtf(d) : 0.f; }
}

__global__ void k_norm(const int* __restrict__ src, const int* __restrict__ dst,
                       const float* __restrict__ w, const float* __restrict__ dinv,
                       float* __restrict__ nrm, int E) {
    int e = blockIdx.x * blockDim.x + threadIdx.x;
    if (e < E) nrm[e] = dinv[src[e]] * w[e] * dinv[dst[e]];
}

// ---------------------------------------------------------------------------
// Layer 0 linear: K = 9, scalar (tiny FLOPs; W0 broadcast-cached)
// ---------------------------------------------------------------------------
__global__ void k_gemm9(const float* __restrict__ X, const float* __restrict__ W,
                        float* __restrict__ Y, int nrows) {
    long idx = (long)blockIdx.x * blockDim.x + threadIdx.x;
    if (idx >= (long)nrows * H) return;
    long n = idx >> 7; int j = (int)(idx & 127);
    const float* xr = X + n * 9;
    float acc = 0.f;
#pragma unroll
    for (int k = 0; k < 9; ++k) acc = fmaf(xr[k], W[k * H + j], acc);
    Y[idx] = acc;
}

// ---------------------------------------------------------------------------
// Message passing: out = selfloop + scatter-add of norm*h[src]
// ---------------------------------------------------------------------------
__global__ void k_selfmsg(const float* __restrict__ dinv,
                          const float* __restrict__ hlin,
                          float* __restrict__ out, int nrows) {
    long idx = (long)blockIdx.x * blockDim.x + threadIdx.x;
    if (idx >= (long)nrows * H) return;
    long n = idx >> 7;
    float di = dinv[n];
    out[idx] = di * di * hlin[idx];
}

__global__ void k_edgemsg(const int* __restrict__ src, const int* __restrict__ dst,
                          const float* __restrict__ nrm,
                          const float* __restrict__ hlin,
                          float* __restrict__ out, int E) {
    long idx = (long)blockIdx.x * blockDim.x + threadIdx.x;
    long e = idx >> 7;
    if (e >= E) return;
    int j = (int)(idx & 127);
    float v = nrm[e] * hlin[(long)src[e] * H + j];
    atomicAdd(out + (long)dst[e] * H + j, v);
}

__global__ void k_bias_relu(float* __restrict__ h, const float* __restrict__ b,
                            int nrows) {
    long idx = (long)blockIdx.x * blockDim.x + threadIdx.x;
    if (idx >= (long)nrows * H) return;
    float v = h[idx] + b[idx & 127];
    h[idx] = v > 0.f ? v : 0.f;
}

// ---------------------------------------------------------------------------
// Pooling
// ---------------------------------------------------------------------------
__global__ void k_zero(float* __restrict__ p, long n) {
    long i = (long)blockIdx.x * blockDim.x + threadIdx.x;
    if (i < n) p[i] = 0.f;
}

__global__ void k_pool_accum(const float* __restrict__ h, const int* __restrict__ batch,
                             float* __restrict__ sums, float* __restrict__ cnt,
                             int nrows) {
    long idx = (long)blockIdx.x * blockDim.x + threadIdx.x;
    if (idx >= (long)nrows * H) return;
    long n = idx >> 7; int j = (int)(idx & 127);
    int b = batch[n];
    atomicAdd(&sums[(long)b * H + j], h[idx]);
    if (j == 0) atomicAdd(&cnt[b], 1.0f);
}

__global__ void k_pool_div(const float* __restrict__ sums, const float* __restrict__ cnt,
                           float* __restrict__ gx, int G) {
    long idx = (long)blockIdx.x * blockDim.x + threadIdx.x;
    if (idx >= (long)G * H) return;
    long g = idx >> 7;
    gx[idx] = sums[idx] / fmaxf(cnt[g], 1.0f);
}

// ---------------------------------------------------------------------------
// Final 128 -> 2 projection
// ---------------------------------------------------------------------------
__global__ void k_ffn2(const float* __restrict__ t1, const float* __restrict__ W,
                       const float* __restrict__ b, float* __restrict__ out, int G) {
    int idx = blockIdx.x * blockDim.x + threadIdx.x;
    if (idx >= G * 2) return;
    int g = idx >> 1, c = idx & 1;
    const float* tr = t1 + (long)g * H;
    float acc = b[c];
#pragma unroll 8
    for (int j = 0; j < H; ++j) acc = fmaf(tr[j], W[j * 2 + c], acc);
    out[idx] = acc;
}

// ---------------------------------------------------------------------------
static inline int cdiv(long a, long b) { return (int)((a + b - 1) / b); }

extern "C" void kernel_launch(void* const* d_in, const int* in_sizes, int n_in,
                              void* d_out, int out_size, void* d_ws, size_t ws_size,
                              hipStream_t stream) {
    const float* x   = (const float*)d_in[0];
    const int*   ei  = (const int*)  d_in[1];
    const float* ea  = (const float*)d_in[2];
    const int*   bat = (const int*)  d_in[3];
    const float* W0  = (const float*)d_in[4];  const float* b0  = (const float*)d_in[5];
    const float* W1  = (const float*)d_in[6];  const float* b1  = (const float*)d_in[7];
    const float* W2  = (const float*)d_in[8];  const float* b2  = (const float*)d_in[9];
    const float* Wf1 = (const float*)d_in[10]; const float* bf1 = (const float*)d_in[11];
    const float* Wf2 = (const float*)d_in[12]; const float* bf2 = (const float*)d_in[13];
    float* out = (float*)d_out;

    const int N = in_sizes[0] / 9;
    const int E = in_sizes[1] / 2;
    const int G = out_size / 2;
    const int* srcp = ei;
    const int* dstp = ei + E;

    // workspace layout
    char* ws = (char*)d_ws;
    size_t off = 0;
    auto take = [&](size_t bytes) -> float* {
        float* p = (float*)(ws + off);
        off += (bytes + 255) & ~(size_t)255;
        return p;
    };
    float* deg  = take((size_t)N * 4);
    float* dinv = take((size_t)N * 4);
    float* nrm  = take((size_t)E * 4);
    float* bufA = take((size_t)N * H * 4);
    float* bufB = take((size_t)N * H * 4);
    float* sums = take((size_t)G * H * 4);
    float* cnt  = take((size_t)G * 4);
    float* gx   = take((size_t)G * H * 4);
    float* t1   = take((size_t)G * H * 4);
    (void)ws_size; (void)n_in;

    const int T = 256;
    const long NH = (long)N * H;
    const long EH = (long)E * H;
    const long GH = (long)G * H;

    // gcn_norm
    k_fill1    <<<cdiv(N, T), T, 0, stream>>>(deg, N);
    k_deg_accum<<<cdiv(E, T), T, 0, stream>>>(dstp, ea, deg, E);
    k_dinv     <<<cdiv(N, T), T, 0, stream>>>(deg, dinv, N);
    k_norm     <<<cdiv(E, T), T, 0, stream>>>(srcp, dstp, ea, dinv, nrm, E);

    // layer 0 (K=9 scalar linear)
    k_gemm9    <<<cdiv(NH, T), T, 0, stream>>>(x, W0, bufA, N);
    k_selfmsg  <<<cdiv(NH, T), T, 0, stream>>>(dinv, bufA, bufB, N);
    k_edgemsg  <<<cdiv(EH, T), T, 0, stream>>>(srcp, dstp, nrm, bufA, bufB, E);
    k_bias_relu<<<cdiv(NH, T), T, 0, stream>>>(bufB, b0, N);

    // layer 1 (WMMA GEMM)
    k_gemm128_wmma<<<cdiv(N, TILE_M), T, 0, stream>>>(bufB, W1, bufA, N);
    k_selfmsg  <<<cdiv(NH, T), T, 0, stream>>>(dinv, bufA, bufB, N);
    k_edgemsg  <<<cdiv(EH, T), T, 0, stream>>>(srcp, dstp, nrm, bufA, bufB, E);
    k_bias_relu<<<cdiv(NH, T), T, 0, stream>>>(bufB, b1, N);

    // layer 2 (WMMA GEMM)
    k_gemm128_wmma<<<cdiv(N, TILE_M), T, 0, stream>>>(bufB, W2, bufA, N);
    k_selfmsg  <<<cdiv(NH, T), T, 0, stream>>>(dinv, bufA, bufB, N);
    k_edgemsg  <<<cdiv(EH, T), T, 0, stream>>>(srcp, dstp, nrm, bufA, bufB, E);
    k_bias_relu<<<cdiv(NH, T), T, 0, stream>>>(bufB, b2, N);

    // global mean pool
    k_zero      <<<cdiv(GH, T), T, 0, stream>>>(sums, GH);
    k_zero      <<<cdiv(G, T),  T, 0, stream>>>(cnt, G);
    k_pool_accum<<<cdiv(NH, T), T, 0, stream>>>(bufB, bat, sums, cnt, N);
    k_pool_div  <<<cdiv(GH, T), T, 0, stream>>>(sums, cnt, gx, G);

    // FFN: relu(gx @ Wf1 + bf1) @ Wf2 + bf2
    k_gemm128_wmma<<<cdiv(G, TILE_M), T, 0, stream>>>(gx, Wf1, t1, G);
    k_bias_relu <<<cdiv(GH, T), T, 0, stream>>>(t1, bf1, G);
    k_ffn2      <<<cdiv((long)G * 2, T), T, 0, stream>>>(t1, Wf2, bf2, out, G);
}